// VectorQuantizer_37967510897188
// MI455X (gfx1250) — compile-verified
//
#include <hip/hip_runtime.h>
#include <hip/hip_bf16.h>

#define C_DIM 256
#define K_CODES 8192
#define N_TOK 32768
#define SPEED 0.01f

typedef __attribute__((ext_vector_type(2))) float v2f;
typedef __attribute__((ext_vector_type(8))) float v8f;

// ---------------------------------------------------------------------------
// Kernel 1: codebook = ces / clip(usage, 1e-5); pre-scale EMA outputs by (1-s)
// ---------------------------------------------------------------------------
__global__ void vq_init(const float* __restrict__ usage,
                        const float* __restrict__ ces,
                        float* __restrict__ cb,
                        float* __restrict__ out_usage,
                        float* __restrict__ out_ces,
                        float* __restrict__ out_loss) {
    size_t idx = (size_t)blockIdx.x * blockDim.x + threadIdx.x;   // over K*C
    int k = (int)(idx >> 8);                                      // C == 256
    float u = usage[k];
    float den = fmaxf(u, 1e-5f);
    float v = ces[idx];
    cb[idx] = v / den;
    out_ces[idx] = v * (1.0f - SPEED);
    if ((idx & 255) == 0) out_usage[k] = u * (1.0f - SPEED);
    if (idx == 0) *out_loss = 0.0f;
}

// ---------------------------------------------------------------------------
// Kernel 2: cbsq[k] = sum_c cb[k][c]^2   (one 256-thread block per code row)
// ---------------------------------------------------------------------------
__global__ void vq_cbsq(const float* __restrict__ cb, float* __restrict__ cbsq) {
    __shared__ float red[256];
    int k = blockIdx.x, c = threadIdx.x;
    float v = cb[(size_t)k * C_DIM + c];
    red[c] = v * v;
    __syncthreads();
    for (int s = 128; s > 0; s >>= 1) {
        if (c < s) red[c] += red[c + s];
        __syncthreads();
    }
    if (c == 0) cbsq[k] = red[0];
}

// ---------------------------------------------------------------------------
// Kernel 3: fused GEMM (f32 WMMA) + argmin over codes.
// 256 threads = 8 waves; 128 tokens per block (16 per wave).
// Code tiles (64 codes x 256 ch) double-buffered in LDS, filled with
// GLOBAL_LOAD_ASYNC_TO_LDS_B128 (ASYNCcnt) while WMMA runs on the prior tile.
// Inner loop software-pipelined: fragments for step i+1 loaded before the
// 4 WMMAs of step i issue, hiding LDS latency.
// ---------------------------------------------------------------------------
#define TOK_PER_BLOCK 128
#define CODE_TILE 64
#define LDS_STRIDE 260   // 256 + 4 pad: keeps 16B alignment, 16 rows -> 16 banks
#define N_TILES (K_CODES / CODE_TILE)

// Issue one async b128 global->LDS copy (GVS addressing).
// ldsOff: byte offset into the wave's LDS allocation (dynamic LDS starts at 0).
// gOff:   32-bit byte offset added to the SGPR64 base.
__device__ __forceinline__ void async_copy_b128(unsigned ldsOff, unsigned gOff,
                                                unsigned long long gBase) {
    asm volatile("global_load_async_to_lds_b128 %0, %1, %2"
                 :: "v"(ldsOff), "v"(gOff), "s"(gBase)
                 : "memory");
}

__launch_bounds__(256)
__global__ void vq_argmin(const float* __restrict__ emb,
                          const float* __restrict__ cb,
                          const float* __restrict__ cbsq,
                          int* __restrict__ codes,
                          float* __restrict__ out_codes) {
    extern __shared__ float lds[];
    const unsigned TOK_FLOATS  = TOK_PER_BLOCK * LDS_STRIDE;  // token tile
    const unsigned CODE_FLOATS = CODE_TILE * LDS_STRIDE;      // one code buffer

    const int tid  = threadIdx.x;
    const int lane = tid & 31;
    const int wave = tid >> 5;
    const int tokBase = blockIdx.x * TOK_PER_BLOCK;

    const unsigned long long embBase =
        (unsigned long long)(emb + (size_t)tokBase * C_DIM);
    const unsigned long long cbBase = (unsigned long long)cb;

    // --- prologue: async-load token tile (128x256) ---------------------------
    // 8192 b128 chunks -> 32 per thread (ASYNCcnt += 32 per wave)
#pragma unroll
    for (int j = 0; j < 32; ++j) {
        int i = tid + j * 256;
        int row = i >> 6, cc = i & 63;               // 64 b128 chunks per row
        unsigned ldsOff = (unsigned)(row * LDS_STRIDE + cc * 4) * 4u;
        unsigned gOff   = (unsigned)(row * C_DIM + cc * 4) * 4u;
        async_copy_b128(ldsOff, gOff, embBase);
    }
    // --- prologue: async-load code tile 0 into buffer 0 ----------------------
#pragma unroll
    for (int j = 0; j < 16; ++j) {
        int i = tid + j * 256;                        // 4096 chunks
        int row = i >> 6, cc = i & 63;
        unsigned ldsOff = (TOK_FLOATS + (unsigned)(row * LDS_STRIDE + cc * 4)) * 4u;
        unsigned gOff   = (unsigned)(row * C_DIM + cc * 4) * 4u;
        async_copy_b128(ldsOff, gOff, cbBase);
    }

    float best[8];
    int   bidx[8];
#pragma unroll
    for (int r = 0; r < 8; ++r) { best[r] = 3.0e38f; bidx[r] = 0; }

    const int kb = (lane >> 4) << 1;   // A/B frag K-subsel: 0 (lanes 0-15) / 2
    const int hl = lane & 15;
    const unsigned tokRow = (unsigned)(wave * 16 + hl) * LDS_STRIDE;

    for (int it = 0; it < N_TILES; ++it) {
        // all waves done computing on the buffer we are about to overwrite
        __syncthreads();
        if (it + 1 < N_TILES) {
            const int k0n = (it + 1) * CODE_TILE;
            const unsigned bufB = TOK_FLOATS + (unsigned)((it + 1) & 1) * CODE_FLOATS;
#pragma unroll
            for (int j = 0; j < 16; ++j) {
                int i = tid + j * 256;
                int row = i >> 6, cc = i & 63;
                unsigned ldsOff = (bufB + (unsigned)(row * LDS_STRIDE + cc * 4)) * 4u;
                unsigned gOff   = (unsigned)((k0n + row) * C_DIM + cc * 4) * 4u;
                async_copy_b128(ldsOff, gOff, cbBase);
            }
            // 16 newest copies (tile it+1) may remain in flight; tile it done
            asm volatile("s_wait_asynccnt 0x10" ::: "memory");
        } else {
            asm volatile("s_wait_asynccnt 0x0" ::: "memory");
        }
        __syncthreads();   // every wave's slice of the current buffer is visible

        const float* tptr = lds + tokRow;
        const float* cbuf = lds + TOK_FLOATS + (unsigned)(it & 1) * CODE_FLOATS;
        const float* c0p  = cbuf + (hl +  0) * LDS_STRIDE;
        const float* c1p  = cbuf + (hl + 16) * LDS_STRIDE;
        const float* c2p  = cbuf + (hl + 32) * LDS_STRIDE;
        const float* c3p  = cbuf + (hl + 48) * LDS_STRIDE;

        v8f acc0 = {}, acc1 = {}, acc2 = {}, acc3 = {};

        // software pipeline: preload step-0 fragments
        v2f a, b0, b1, b2, b3;
        a.x  = tptr[kb]; a.y  = tptr[kb + 1];
        b0.x = c0p[kb];  b0.y = c0p[kb + 1];
        b1.x = c1p[kb];  b1.y = c1p[kb + 1];
        b2.x = c2p[kb];  b2.y = c2p[kb + 1];
        b3.x = c3p[kb];  b3.y = c3p[kb + 1];

#pragma unroll 4
        for (int c0 = 0; c0 < C_DIM; c0 += 4) {
            const int cn = ((c0 + 4) & (C_DIM - 1)) + kb;  // wrap: last prefetch dummy
            v2f an, b0n, b1n, b2n, b3n;
            an.x  = tptr[cn]; an.y  = tptr[cn + 1];
            b0n.x = c0p[cn];  b0n.y = c0p[cn + 1];
            b1n.x = c1p[cn];  b1n.y = c1p[cn + 1];
            b2n.x = c2p[cn];  b2n.y = c2p[cn + 1];
            b3n.x = c3p[cn];  b3n.y = c3p[cn + 1];
            acc0 = __builtin_amdgcn_wmma_f32_16x16x4_f32(
                false, a, false, b0, (short)0, acc0, false, false);
            acc1 = __builtin_amdgcn_wmma_f32_16x16x4_f32(
                false, a, false, b1, (short)0, acc1, false, false);
            acc2 = __builtin_amdgcn_wmma_f32_16x16x4_f32(
                false, a, false, b2, (short)0, acc2, false, false);
            acc3 = __builtin_amdgcn_wmma_f32_16x16x4_f32(
                false, a, false, b3, (short)0, acc3, false, false);
            a = an; b0 = b0n; b1 = b1n; b2 = b2n; b3 = b3n;
        }

        // score = ||cb||^2 - 2*dot  (||x||^2 constant per token -> irrelevant)
        const int codeB = it * CODE_TILE + hl;
        const float cs0 = cbsq[codeB];
        const float cs1 = cbsq[codeB + 16];
        const float cs2 = cbsq[codeB + 32];
        const float cs3 = cbsq[codeB + 48];
#pragma unroll
        for (int r = 0; r < 8; ++r) {
            float s0 = cs0 - 2.0f * acc0[r];
            float s1 = cs1 - 2.0f * acc1[r];
            float s2 = cs2 - 2.0f * acc2[r];
            float s3 = cs3 - 2.0f * acc3[r];
            if (s0 < best[r]) { best[r] = s0; bidx[r] = codeB; }
            if (s1 < best[r]) { best[r] = s1; bidx[r] = codeB + 16; }
            if (s2 < best[r]) { best[r] = s2; bidx[r] = codeB + 32; }
            if (s3 < best[r]) { best[r] = s3; bidx[r] = codeB + 48; }
        }
    }

    // cross-lane argmin within each 16-lane half (wave32; offsets<=8 keep halves)
#pragma unroll
    for (int off = 8; off >= 1; off >>= 1) {
#pragma unroll
        for (int r = 0; r < 8; ++r) {
            float ov = __shfl_xor(best[r], off, 32);
            int   oi = __shfl_xor(bidx[r], off, 32);
            if (ov < best[r] || (ov == best[r] && oi < bidx[r])) {
                best[r] = ov;
                bidx[r] = oi;
            }
        }
    }

    if (hl == 0) {
        // D layout: VGPR r holds M=r (lanes 0-15) / M=r+8 (lanes 16-31)
        const int tokSub = (lane >> 4) * 8;
#pragma unroll
        for (int r = 0; r < 8; ++r) {
            int token = tokBase + wave * 16 + tokSub + r;
            codes[token]     = bidx[r];
            out_codes[token] = (float)bidx[r];
        }
    }
}

// ---------------------------------------------------------------------------
// Kernel 4: gather quantized rows, commitment loss, EMA accumulation.
// One 256-thread block per token, one thread per channel.
// ---------------------------------------------------------------------------
__global__ void vq_scatter(const float* __restrict__ emb,
                           const float* __restrict__ cb,
                           const int* __restrict__ codes,
                           float* __restrict__ out_q,
                           float* __restrict__ out_loss,
                           float* __restrict__ out_usage,
                           float* __restrict__ out_ces) {
    __shared__ float red[256];
    const int n = blockIdx.x;
    const int c = threadIdx.x;
    const int code = codes[n];
    const float e = emb[(size_t)n * C_DIM + c];
    const float q = cb[(size_t)code * C_DIM + c];
    out_q[(size_t)n * C_DIM + c] = q;   // embeddings_q == quantized numerically
    const float d = q - e;
    red[c] = d * d;
    atomicAdd(&out_ces[(size_t)code * C_DIM + c], SPEED * e);
    __syncthreads();
    for (int s = 128; s > 0; s >>= 1) {
        if (c < s) red[c] += red[c + s];
        __syncthreads();
    }
    if (c == 0) {
        atomicAdd(out_loss, red[0] * (1.0f / ((float)N_TOK * (float)C_DIM)));
        atomicAdd(&out_usage[code], SPEED);
    }
}

// ---------------------------------------------------------------------------
extern "C" void kernel_launch(void* const* d_in, const int* in_sizes, int n_in,
                              void* d_out, int out_size, void* d_ws, size_t ws_size,
                              hipStream_t stream) {
    const float* emb   = (const float*)d_in[0];  // [N_TOK, C]
    const float* usage = (const float*)d_in[1];  // [K]
    const float* ces   = (const float*)d_in[2];  // [K, C]

    float* ws   = (float*)d_ws;
    float* cb   = ws;                                  // K*C floats
    float* cbsq = ws + (size_t)K_CODES * C_DIM;        // K floats
    int*   code_ws = (int*)(cbsq + K_CODES);           // N_TOK ints

    float* out       = (float*)d_out;
    float* out_codes = out;                                     // [N_TOK]
    float* out_q     = out + N_TOK;                             // [N_TOK*C]
    float* out_loss  = out_q + (size_t)N_TOK * C_DIM;           // [1]
    float* out_usage = out_loss + 1;                            // [K]
    float* out_ces   = out_usage + K_CODES;                     // [K*C]

    vq_init<<<(K_CODES * C_DIM) / 256, 256, 0, stream>>>(
        usage, ces, cb, out_usage, out_ces, out_loss);

    vq_cbsq<<<K_CODES, 256, 0, stream>>>(cb, cbsq);

    const size_t lds_bytes =
        (size_t)(TOK_PER_BLOCK + 2 * CODE_TILE) * LDS_STRIDE * sizeof(float);
    vq_argmin<<<N_TOK / TOK_PER_BLOCK, 256, lds_bytes, stream>>>(
        emb, cb, cbsq, code_ws, out_codes);

    vq_scatter<<<N_TOK, 256, 0, stream>>>(
        emb, cb, code_ws, out_q, out_loss, out_usage, out_ces);
}